// PINNPhysicsLoss_9268539424854
// MI455X (gfx1250) — compile-verified
//
#include <hip/hip_runtime.h>
#include <hip/hip_bf16.h>

#define BDIM   256
#define W      256
#define H      256
#define TT     32
#define NB     16
#define TY     16
#define NYT    (H / TY)            // 16 y-tiles
#define NTM1   (TT - 1)            // 31
#define NBLK   (NB * NTM1 * NYT)   // 7936 blocks / partials
#define C_LIGHT 300000000.0f
#define HDELTA  0.1f
#define NPHYS   32505856.0f        // 16*31*256*256

typedef int v4i __attribute__((vector_size(16)));                 // matches builtin param
typedef __attribute__((address_space(1))) v4i gv4i;               // global (AS1)
typedef __attribute__((address_space(3))) v4i lv4i;               // LDS (AS3)
typedef float v2f __attribute__((ext_vector_type(2)));
typedef float v8f __attribute__((ext_vector_type(8)));

__device__ __forceinline__ float huber(float x) {
    float a = fabsf(x);
    return (a <= HDELTA) ? 0.5f * x * x : HDELTA * (a - 0.5f * HDELTA);
}

// ---- CDNA5 async global->LDS staging (ASYNCcnt path) ----------------------
__device__ __forceinline__ void async_ld_b128(const float* g, float* l) {
#if __has_builtin(__builtin_amdgcn_global_load_async_to_lds_b128)
    __builtin_amdgcn_global_load_async_to_lds_b128(
        (gv4i*)(unsigned long long)(const void*)g,   // flat==global numerically
        (lv4i*)l,                                    // addrspacecast: flat -> LDS
        0, 0);
#else
    unsigned loff = (unsigned)(unsigned long long)(lv4i*)l;
    unsigned long long ga = (unsigned long long)(const void*)g;
    asm volatile("global_load_async_to_lds_b128 %0, %1, off"
                 :: "v"(loff), "v"(ga) : "memory");
#endif
}

__device__ __forceinline__ void wait_async0() {
#if __has_builtin(__builtin_amdgcn_s_wait_asynccnt)
    __builtin_amdgcn_s_wait_asynccnt(0);
#else
    asm volatile("s_wait_asynccnt 0" ::: "memory");
#endif
}

// ---- stencil + Huber + block reduction ------------------------------------
__global__ __launch_bounds__(BDIM) void physics_kernel(
    const float* __restrict__ phi,
    const float* __restrict__ n_pred,
    const float* __restrict__ mu_a_p,
    const float* __restrict__ mu_s_p,
    float* __restrict__ partials)
{
    __shared__ __align__(16) float tile[20 * W];   // rows y0-2 .. y0+17 of phi[b,t]
    __shared__ float wsum[BDIM / 32];

    const int tid = threadIdx.x;
    const int bid = blockIdx.x;
    const int yt  = bid & (NYT - 1);
    const int t   = (bid / NYT) % NTM1;
    const int b   = bid / (NYT * NTM1);
    const int y0  = yt * TY;

    const float mu_a = mu_a_p[b];
    const float Dc   = 1.0f / (3.0f * (mu_a + mu_s_p[b]));
    const float nC   = n_pred[b] / C_LIGHT;

    const size_t slice = (size_t)(b * TT + t) * (size_t)(H * W);

    // Stage 20 rows (2-row halo top/bottom) via async b128 copies.
    // r_local = g*4 + tid/64 is wave-uniform -> uniform branches.
    {
        const int rl = tid >> 6;            // 0..3
        const int x4 = (tid & 63) * 4;      // 16B-aligned column start
        #pragma unroll
        for (int g = 0; g < 5; ++g) {
            const int r  = g * 4 + rl;      // 0..19
            const int gy = y0 - 2 + r;
            if (gy >= 0 && gy < H)
                async_ld_b128(phi + slice + (size_t)gy * W + x4, &tile[r * W + x4]);
        }
    }
    wait_async0();
    __syncthreads();

    const int x = tid;                       // one column per thread
    const float* __restrict__ phin = phi + slice + (size_t)(H * W);  // t+1 slice
    float acc = 0.0f;

    #pragma unroll 4
    for (int j = 0; j < TY; ++j) {
        const int y  = y0 + j;
        const int ry = j + 2;
        const float pc = tile[ry * W + x];

        // d/dx( cdx ) with zero-padded boundaries, collapsed:
        float lapx;
        if (x == 0 || x == W - 1)  lapx = 0.0f;
        else if (x == 1)           lapx =  (tile[ry * W + 3]     - pc) * 0.25f;
        else if (x == W - 2)       lapx = -(pc - tile[ry * W + W - 4]) * 0.25f;
        else                       lapx =  (tile[ry * W + x + 2] - 2.0f * pc
                                          + tile[ry * W + x - 2]) * 0.25f;
        float lapy;
        if (y == 0 || y == H - 1)  lapy = 0.0f;
        else if (y == 1)           lapy =  (tile[(ry + 2) * W + x] - pc) * 0.25f;
        else if (y == H - 2)       lapy = -(pc - tile[(ry - 2) * W + x]) * 0.25f;
        else                       lapy =  (tile[(ry + 2) * W + x] - 2.0f * pc
                                          + tile[(ry - 2) * W + x]) * 0.25f;

        const float dphi = phin[(size_t)y * W + x] - pc;   // DT = 1
        const float r = nC * dphi + mu_a * pc - Dc * (lapx + lapy);
        acc += huber(r);
    }

    // deterministic wave32 tree + cross-wave reduce
    for (int off = 16; off > 0; off >>= 1)
        acc += __shfl_down(acc, off, 32);
    if ((tid & 31) == 0) wsum[tid >> 5] = acc;
    __syncthreads();
    if (tid == 0) {
        float s = 0.0f;
        #pragma unroll
        for (int w = 0; w < BDIM / 32; ++w) s += wsum[w];
        partials[bid] = s;
    }
}

// ---- final reduction (one wave); 32->1 via V_WMMA_F32_16X16X4_F32 ---------
__global__ __launch_bounds__(32) void finalize_kernel(
    const float* __restrict__ partials,
    const float* __restrict__ n_pred,
    const float* __restrict__ n_true,
    float* __restrict__ out)
{
    const int lane = threadIdx.x;

    float p = 0.0f;
    for (int i = lane; i < NBLK; i += 32)   // 248 iters per lane, fixed order
        p += partials[i];

    float tot;
#if __has_builtin(__builtin_amdgcn_wmma_f32_16x16x4_f32)
    // A = ones(16x4), B holds the 32 lane partials (rest 0).
    // D[m][n] = sum_k B[k][n]  => every D element of column n is that column's
    // sum; summing D row 0 over the 16 columns gives the exact 32-lane total.
    v2f a;  a[0] = 1.0f; a[1] = 1.0f;
    v2f bm; bm[0] = p;   bm[1] = 0.0f;
    v8f c = {};
    c = __builtin_amdgcn_wmma_f32_16x16x4_f32(false, a, false, bm,
                                              (short)0, c, false, false);
    float cs = c[0];
    cs += __shfl_down(cs, 8, 16);
    cs += __shfl_down(cs, 4, 16);
    cs += __shfl_down(cs, 2, 16);
    cs += __shfl_down(cs, 1, 16);
    tot = cs;                                // valid on lane 0
#else
    for (int off = 16; off > 0; off >>= 1)
        p += __shfl_down(p, off, 32);
    tot = p;
#endif

    float hd = (lane < 16) ? huber(n_pred[lane] - n_true[lane]) : 0.0f;
    for (int off = 16; off > 0; off >>= 1)
        hd += __shfl_down(hd, off, 32);

    if (lane == 0) {
        const float physics = tot / NPHYS;
        const float data    = hd / 16.0f;
        out[0] = data + physics;
        out[1] = data;
        out[2] = physics;
    }
}

extern "C" void kernel_launch(void* const* d_in, const int* in_sizes, int n_in,
                              void* d_out, int out_size, void* d_ws, size_t ws_size,
                              hipStream_t stream) {
    const float* phi    = (const float*)d_in[0];
    const float* npred  = (const float*)d_in[1];
    const float* ntrue  = (const float*)d_in[2];
    const float* mua    = (const float*)d_in[3];
    const float* mus    = (const float*)d_in[4];
    float* out      = (float*)d_out;
    float* partials = (float*)d_ws;    // NBLK floats, fully rewritten each call

    physics_kernel<<<NBLK, BDIM, 0, stream>>>(phi, npred, mua, mus, partials);
    finalize_kernel<<<1, 32, 0, stream>>>(partials, npred, ntrue, out);
}